// EnsembleGILRLSTMLayer_67980742362010
// MI455X (gfx1250) — compile-verified
//
#include <hip/hip_runtime.h>
#include <stdint.h>

typedef float v2f __attribute__((ext_vector_type(2)));
typedef float v8f __attribute__((ext_vector_type(8)));
typedef unsigned int v4u __attribute__((ext_vector_type(4)));
typedef int v4i __attribute__((ext_vector_type(4)));
typedef int v8i __attribute__((ext_vector_type(8)));

#define E_    8
#define B_    16
#define T_    512
#define DDIM  256
#define ROWS_ (B_ * T_)    // 8192 rows per ensemble member
#define KDIM  256
#define NDIM  256

#ifndef __has_builtin
#define __has_builtin(x) 0
#endif
#if __has_builtin(__builtin_amdgcn_tensor_load_to_lds) && __has_builtin(__builtin_amdgcn_s_wait_tensorcnt)
#define USE_TDM 1
#else
#define USE_TDM 0
#endif

__device__ __forceinline__ float sigmoidf_(float x) {
    return 1.0f / (1.0f + __expf(-x));
}

#if USE_TDM
// Issue one TDM 2D tile load Global->LDS.  tile_w elems per row, tile_h rows,
// global row stride row_stride_elems, LDS rows padded by (pad_amount_code+1)
// DWORDs every 2^(pad_interval_code+1) DWORDs.  data_size = 4 bytes.
__device__ __forceinline__ void tdm_load_2d(unsigned lds_off, const float* gptr,
                                            unsigned tile_w, unsigned tile_h,
                                            unsigned row_stride_elems,
                                            unsigned pad_interval_code,
                                            unsigned pad_amount_code) {
    const unsigned long long ga = (unsigned long long)(uintptr_t)gptr;
    v4u g0;
    g0[0] = 1u;                                                  // count=1, user-mode D#
    g0[1] = lds_off;                                             // lds_addr (bytes)
    g0[2] = (unsigned)ga;                                        // global_addr[31:0]
    g0[3] = (unsigned)((ga >> 32) & 0x01FFFFFFull) | (2u << 30); // addr[56:32] | type=2
    v8i g1;
    g1[0] = (int)((2u << 16)                      // data_size = 4B
                  | (1u << 20)                    // pad_enable
                  | (pad_interval_code << 22)
                  | (pad_amount_code << 25));
    g1[1] = (int)((tile_w & 0xFFFFu) << 16);                       // tensor_dim0[15:0]
    g1[2] = (int)((tile_w >> 16) | ((tile_h & 0xFFFFu) << 16));    // td0 hi | td1 lo
    g1[3] = (int)((tile_h >> 16) | (tile_w << 16));                // td1 hi | tile_dim0
    g1[4] = (int)(tile_h & 0xFFFFu);                               // tile_dim1 (tile_dim2=0)
    g1[5] = (int)row_stride_elems;                                 // tensor_dim0_stride[31:0]
    g1[6] = 0;
    g1[7] = 0;
    const v4i z4 = {0, 0, 0, 0};
    const v8i z8 = {0, 0, 0, 0, 0, 0, 0, 0};
    __builtin_amdgcn_tensor_load_to_lds(g0, g1, z4, z4, z8, 0);
}
#endif

// ---------------------------------------------------------------------------
// Weight pre-transpose: dst[mat][n][k] = src[mat][k][n] for 256x256 matrices.
// Makes WMMA B-fragments (B[k][n], B[k+1][n]) contiguous in LDS tiles.
// Grid: (8, 8, nmat); 256 threads; 32x32 LDS tile (+1 pad).
// ---------------------------------------------------------------------------
__global__ __launch_bounds__(256)
void transpose_kernel(const float* __restrict__ src, float* __restrict__ dst) {
    __shared__ float t[32][33];
    const size_t mat = blockIdx.z;
    const float* s = src + mat * (size_t)(KDIM * NDIM);
    float*       d = dst + mat * (size_t)(KDIM * NDIM);
    const int x  = blockIdx.x * 32;           // col base in src
    const int y  = blockIdx.y * 32;           // row base in src
    const int tx = threadIdx.x & 31;
    const int ty = threadIdx.x >> 5;          // 8 rows per pass
#pragma unroll
    for (int j = 0; j < 32; j += 8)
        t[ty + j][tx] = s[(size_t)(y + ty + j) * NDIM + x + tx];
    __syncthreads();
#pragma unroll
    for (int j = 0; j < 32; j += 8)
        d[(size_t)(x + ty + j) * KDIM + y + tx] = t[tx][ty + j];
}

// ---------------------------------------------------------------------------
// Fused multi-matrix GEMM + activation epilogue.
//   A:  [E][ROWS][K] row-major
//   Wt: [NM][E][N][K]  (pre-transposed weights)
//   bias: [NM][E][N], rs: [ROWS] or null.
// MODE 0 (NM=2): out0 = tanh(u0+b0); out1 = sigmoid(u1+b1)*(1-rs)
// MODE 1 (NM=4): out0 = sigmoid(u0+b0)*(1-rs); out1 = sigmoid(u1+b1)*tanh(u3+b3);
//                out2 = sigmoid(u2+b2)
// MODE 2 (NM=1): out0 = u0 + b0
// Block: 256 threads (8 wave32) -> one 64x64 output tile; wave w owns subtile
// row w>>1 and subtile cols {(w&1)*2, (w&1)*2+1} (A fragment shared).
// A and B LDS tiles share the same padded [64][ST] layout -> both fragments
// are aligned v2f (ds_load_b64) reads, conflict-free across 64 banks.
// ---------------------------------------------------------------------------
template<int NM, int MODE>
__global__ __launch_bounds__(256)
void gemm_act_kernel(const float* __restrict__ A,
                     const float* __restrict__ Wt,
                     const float* __restrict__ bias,
                     const float* __restrict__ rs,
                     float* __restrict__ out0,
                     float* __restrict__ out1,
                     float* __restrict__ out2) {
    constexpr int KC   = (NM == 4) ? 16 : 32;   // k-chunk
    constexpr int AST  = (NM == 4) ? 20 : 34;   // LDS row stride (pad -> 64-bank clean)
    constexpr int APIC = (NM == 4) ? 3  : 4;    // pad interval code: 2^(c+1) dwords = KC
    constexpr int APAC = (NM == 4) ? 3  : 1;    // pad amount code: (c+1) dwords
    constexpr int TBUF = 64 * AST;              // floats per (A or B) tile buffer

    __shared__ float As[2][TBUF];
    __shared__ float Bs[2][NM][TBUF];

    const int e    = blockIdx.z;
    const int r0   = blockIdx.x * 64;
    const int c0   = blockIdx.y * 64;
    const int tid  = threadIdx.x;
    const int lane = tid & 31;
    const int wv   = tid >> 5;
    const int sr   = wv >> 1;
    const int sc0  = (wv & 1) * 2;
    const int lo16 = lane & 15;
    const int hi   = lane >> 4;

    const float* Ag = A + ((size_t)e * ROWS_ + r0) * KDIM;

    v8f acc[NM][2];
#pragma unroll
    for (int m = 0; m < NM; ++m)
#pragma unroll
        for (int c = 0; c < 2; ++c)
#pragma unroll
            for (int i = 0; i < 8; ++i) acc[m][c][i] = 0.0f;

#if USE_TDM
    const unsigned ldsA = (unsigned)(unsigned long long)(uintptr_t)&As[0][0];
    const unsigned ldsB = (unsigned)(unsigned long long)(uintptr_t)&Bs[0][0][0];
    // ---- preload k-block 0 into buffer 0 (wave 0 issues DMA) ----
    if (wv == 0) {
        tdm_load_2d(ldsA, Ag, KC, 64, KDIM, APIC, APAC);
#pragma unroll
        for (int m = 0; m < NM; ++m)
            tdm_load_2d(ldsB + (unsigned)(m * TBUF) * 4u,
                        Wt + ((size_t)m * E_ + e) * KDIM * NDIM + (size_t)c0 * KDIM,
                        KC, 64, KDIM, APIC, APAC);
        __builtin_amdgcn_s_wait_tensorcnt(0);
    }
    __syncthreads();
#else
    // ---- manual fallback staging of buffer `b`, k-offset k0 ----
    auto stage = [&](int b, int k0) {
        if constexpr (KC == 32) {
#pragma unroll
            for (int it = 0; it < 2; ++it) {
                const int idx = tid * 2 + it;          // 0..511
                const int row = idx >> 3, c4 = (idx & 7) * 4;
                const float4 va = *(const float4*)(Ag + (size_t)row * KDIM + k0 + c4);
                float* p = &As[b][row * AST + c4];
                p[0] = va.x; p[1] = va.y; p[2] = va.z; p[3] = va.w;
            }
        } else {
            const int row = tid >> 2, c4 = (tid & 3) * 4;
            const float4 va = *(const float4*)(Ag + (size_t)row * KDIM + k0 + c4);
            float* p = &As[b][row * AST + c4];
            p[0] = va.x; p[1] = va.y; p[2] = va.z; p[3] = va.w;
        }
#pragma unroll
        for (int m = 0; m < NM; ++m) {
            const float* Wm = Wt + ((size_t)m * E_ + e) * KDIM * NDIM + (size_t)c0 * KDIM;
            if constexpr (KC == 32) {
#pragma unroll
                for (int it = 0; it < 2; ++it) {
                    const int idx = tid * 2 + it;
                    const int row = idx >> 3, c4 = (idx & 7) * 4;
                    const float4 vb = *(const float4*)(Wm + (size_t)row * KDIM + k0 + c4);
                    float* p = &Bs[b][m][row * AST + c4];
                    p[0] = vb.x; p[1] = vb.y; p[2] = vb.z; p[3] = vb.w;
                }
            } else {
                const int row = tid >> 2, c4 = (tid & 3) * 4;
                const float4 vb = *(const float4*)(Wm + (size_t)row * KDIM + k0 + c4);
                float* p = &Bs[b][m][row * AST + c4];
                p[0] = vb.x; p[1] = vb.y; p[2] = vb.z; p[3] = vb.w;
            }
        }
    };
    stage(0, 0);
    __syncthreads();
#endif

    int buf = 0;
    for (int k0 = 0; k0 < KDIM; k0 += KC) {
        const int nb = buf ^ 1;
        const bool has_next = (k0 + KC) < KDIM;
#if USE_TDM
        if (has_next && wv == 0) {
            tdm_load_2d(ldsA + (unsigned)(nb * TBUF) * 4u, Ag + k0 + KC,
                        KC, 64, KDIM, APIC, APAC);
#pragma unroll
            for (int m = 0; m < NM; ++m)
                tdm_load_2d(ldsB + (unsigned)((nb * NM + m) * TBUF) * 4u,
                            Wt + ((size_t)m * E_ + e) * KDIM * NDIM
                               + (size_t)c0 * KDIM + k0 + KC,
                            KC, 64, KDIM, APIC, APAC);
        }
#else
        if (has_next) stage(nb, k0 + KC);
#endif
        const float* Ab = &As[buf][0];
#pragma unroll
        for (int kk = 0; kk < KC; kk += 4) {
            const int ka = kk + hi * 2;                 // K pair for this half-wave
            const v2f af = *(const v2f*)(Ab + (sr * 16 + lo16) * AST + ka);
#pragma unroll
            for (int m = 0; m < NM; ++m) {
                const float* Bm = &Bs[buf][m][0];
#pragma unroll
                for (int c = 0; c < 2; ++c) {
                    const v2f bf = *(const v2f*)(Bm + ((sc0 + c) * 16 + lo16) * AST + ka);
                    acc[m][c] = __builtin_amdgcn_wmma_f32_16x16x4_f32(
                        false, af, false, bf, (short)0, acc[m][c], false, false);
                }
            }
        }
#if USE_TDM
        if (wv == 0) __builtin_amdgcn_s_wait_tensorcnt(0);
#endif
        __syncthreads();
        buf = nb;
    }

    // ---- epilogue: bias + activations ----
#pragma unroll
    for (int c = 0; c < 2; ++c) {
        const int colg = c0 + (sc0 + c) * 16 + lo16;
        float bmv[NM];
#pragma unroll
        for (int m = 0; m < NM; ++m)
            bmv[m] = bias[((size_t)m * E_ + e) * NDIM + colg];
#pragma unroll
        for (int i = 0; i < 8; ++i) {
            const int rowg = r0 + sr * 16 + hi * 8 + i;   // C/D: VGPR i -> M=i (+8 hi half)
            const size_t oidx = ((size_t)e * ROWS_ + rowg) * (size_t)NDIM + colg;
            if constexpr (MODE == 0) {
                const float g = 1.0f - rs[rowg];
                out0[oidx] = tanhf(acc[0][c][i] + bmv[0]);
                out1[oidx] = sigmoidf_(acc[1][c][i] + bmv[1]) * g;
            } else if constexpr (MODE == 1) {
                const float g  = 1.0f - rs[rowg];
                const float ff = sigmoidf_(acc[0][c][i] + bmv[0]) * g;
                const float ii = sigmoidf_(acc[1][c][i] + bmv[1]);
                const float oo = sigmoidf_(acc[2][c][i] + bmv[2]);
                const float zz = tanhf(acc[3][c][i] + bmv[3]);
                out0[oidx] = ff;
                out1[oidx] = ii * zz;
                out2[oidx] = oo;
            } else {
                out0[oidx] = acc[0][c][i] + bmv[0];
            }
        }
    }
}

// ---------------------------------------------------------------------------
// Scan 1: h_t = f_t*h_{t-1} + (1-f_t)*v_t, in-place over v_ws.
// ---------------------------------------------------------------------------
__global__ __launch_bounds__(256)
void scan1_kernel(float* __restrict__ v_ws, const float* __restrict__ f_ws,
                  const float* __restrict__ hidden, float* __restrict__ hout) {
    const int n = blockIdx.x;
    const int d = threadIdx.x;
    float h = hidden[(size_t)n * (2 * DDIM) + d];
    size_t idx = ((size_t)n * T_) * DDIM + d;
#pragma unroll 4
    for (int t = 0; t < T_; ++t) {
        const float f = f_ws[idx];
        const float v = v_ws[idx];
        h = fmaf(f, h - v, v);       // v + f*(h-v)
        v_ws[idx] = h;
        idx += DDIM;
    }
    hout[(size_t)n * (2 * DDIM) + d] = h;
}

// ---------------------------------------------------------------------------
// Scan 2: h_t = f_t*h_{t-1} + (1-f_t)*(i*z)_t ; store h_t * o_t into iz_ws.
// ---------------------------------------------------------------------------
__global__ __launch_bounds__(256)
void scan2_kernel(const float* __restrict__ f2_ws, float* __restrict__ iz_ws,
                  const float* __restrict__ o_ws,
                  const float* __restrict__ hidden, float* __restrict__ hout) {
    const int n = blockIdx.x;
    const int d = threadIdx.x;
    float h = hidden[(size_t)n * (2 * DDIM) + DDIM + d];
    size_t idx = ((size_t)n * T_) * DDIM + d;
#pragma unroll 4
    for (int t = 0; t < T_; ++t) {
        const float f = f2_ws[idx];
        const float z = iz_ws[idx];
        h = fmaf(f, h - z, z);
        iz_ws[idx] = h * o_ws[idx];
        idx += DDIM;
    }
    hout[(size_t)n * (2 * DDIM) + DDIM + d] = h;
}

// ---------------------------------------------------------------------------
extern "C" void kernel_launch(void* const* d_in, const int* in_sizes, int n_in,
                              void* d_out, int out_size, void* d_ws, size_t ws_size,
                              hipStream_t stream) {
    const float* x      = (const float*)d_in[0];   // [E,B,T,DIN]
    const float* hidden = (const float*)d_in[1];   // [1,E*B,2D]
    const float* rs     = (const float*)d_in[2];   // [B,T,1]
    const float* W_in   = (const float*)d_in[3];   // [2,E,DIN,D]
    const float* b_in   = (const float*)d_in[4];   // [2,E,D]
    const float* W_mid  = (const float*)d_in[5];   // [4,E,D,D]
    const float* b_mid  = (const float*)d_in[6];   // [4,E,D]
    const float* W_out  = (const float*)d_in[7];   // [E,D,D]
    const float* b_out  = (const float*)d_in[8];   // [E,D]
    float* out = (float*)d_out;

    const size_t SZ  = (size_t)E_ * ROWS_ * DDIM;  // 16,777,216 floats (64 MB)
    const size_t MSZ = (size_t)KDIM * NDIM;        // 65,536 floats per matrix
    float* ws     = (float*)d_ws;
    float* v_ws   = ws;            // stage1 v -> scan1 output -> stage2 A
    float* f_ws   = ws + SZ;       // stage1 f (dead after scan1; reused as o_ws)
    float* f2_ws  = ws + 2 * SZ;   // stage2 forget gate
    float* iz_ws  = ws + 3 * SZ;   // stage2 i*z -> scan2 output -> stage3 A
    float* o_ws   = f_ws;          // alias (f_ws dead before stage2 writes)
    float* wt_in  = ws + 4 * SZ;               // [2*E] transposed 256x256
    float* wt_mid = wt_in + 2 * E_ * MSZ;      // [4*E]
    float* wt_out = wt_mid + 4 * E_ * MSZ;     // [E]

    float* hidden_out = out + SZ; // [1, E*B, 2D] tail of d_out

    dim3 blk(256);

    // Weight pre-transposes (14 MB total; negligible vs GEMM work)
    transpose_kernel<<<dim3(8, 8, 2 * E_), blk, 0, stream>>>(W_in,  wt_in);
    transpose_kernel<<<dim3(8, 8, 4 * E_), blk, 0, stream>>>(W_mid, wt_mid);
    transpose_kernel<<<dim3(8, 8, E_),     blk, 0, stream>>>(W_out, wt_out);

    dim3 grid(ROWS_ / 64, NDIM / 64, E_);   // 128 x 4 x 8

    // Stage 1: x @ W_in (m=0,1) -> v = tanh, f = sigmoid * (1-rs)
    gemm_act_kernel<2, 0><<<grid, blk, 0, stream>>>(x, wt_in, b_in, rs, v_ws, f_ws, nullptr);

    // Scan 1 (GILR): in-place on v_ws; h_pre_last -> hidden_out[:, :D]
    scan1_kernel<<<dim3(E_ * B_), blk, 0, stream>>>(v_ws, f_ws, hidden, hidden_out);

    // Stage 2: v4 @ W_mid (m=0..3) -> f2, i*z, o
    gemm_act_kernel<4, 1><<<grid, blk, 0, stream>>>(v_ws, wt_mid, b_mid, rs, f2_ws, iz_ws, o_ws);

    // Scan 2 (LSTM cell): iz_ws <- h*o; h_mid_last -> hidden_out[:, D:]
    scan2_kernel<<<dim3(E_ * B_), blk, 0, stream>>>(f2_ws, iz_ws, o_ws, hidden, hidden_out);

    // Stage 3: out = (h*o) @ W_out + b_out
    gemm_act_kernel<1, 2><<<grid, blk, 0, stream>>>(iz_ws, wt_out, b_out, nullptr, out, nullptr, nullptr);
}